// PR_RoutingLayer_18717467476214
// MI455X (gfx1250) — compile-verified
//
#include <hip/hip_runtime.h>
#include <hip/hip_bf16.h>

#define IN_DIM    128
#define OUT_DIM   128
#define NUM_PERSP 4
#define WCOLS     512   // OUT_DIM * NUM_PERSP
#define CLIP_VAL  10000.0f

typedef float v2f __attribute__((ext_vector_type(2)));
typedef float v8f __attribute__((ext_vector_type(8)));

__device__ __forceinline__ float waveReduceSum(float v) {
    v += __shfl_xor(v, 16, 32);
    v += __shfl_xor(v, 8, 32);
    v += __shfl_xor(v, 4, 32);
    v += __shfl_xor(v, 2, 32);
    v += __shfl_xor(v, 1, 32);
    return v;
}

__device__ __forceinline__ void atomicMaxF(float* addr, float val) {
    // Standard ordered-bit trick: works for mixed signs with -inf init.
    if (!(val < 0.0f))
        atomicMax((int*)addr, __float_as_int(val));
    else
        atomicMin((unsigned int*)addr, __float_as_uint(val));
}

// ---------------------------------------------------------------------------
// 1) Per-node log map. ref = e0 (sqrt(|beta|)=1). One wave per node.
//    inner(x,ref) = -x0 ; ref_sq = -1 ; x_sq = sum(xi^2, i>0) - x0^2
// ---------------------------------------------------------------------------
__global__ __launch_bounds__(256) void tangent_kernel(
    const float* __restrict__ x, float* __restrict__ tangent, int n_nodes) {
    int wid  = (blockIdx.x * blockDim.x + threadIdx.x) >> 5;
    int lane = threadIdx.x & 31;
    if (wid >= n_nodes) return;
    int d0 = lane * 4;
    float4 v = *(const float4*)(x + (size_t)wid * IN_DIM + d0);
    v.x = fminf(fmaxf(v.x, -CLIP_VAL), CLIP_VAL);
    v.y = fminf(fmaxf(v.y, -CLIP_VAL), CLIP_VAL);
    v.z = fminf(fmaxf(v.z, -CLIP_VAL), CLIP_VAL);
    v.w = fminf(fmaxf(v.w, -CLIP_VAL), CLIP_VAL);
    float ss = v.x * v.x + v.y * v.y + v.z * v.z + v.w * v.w;
    ss = waveReduceSum(ss);
    float x0   = __shfl(v.x, 0, 32);           // clipped x[node,0]
    float x_sq = ss - 2.0f * x0 * x0;          // -x0^2 + sum(rest^2)
    float denom = sqrtf(fmaxf(fabsf(x_sq), 1e-12f));
    float cosd  = fmaxf(x0 / denom, 1.0f + 1e-6f);
    float s2    = sqrtf(cosd * cosd - 1.0f);
    float dist  = logf(cosd + s2);             // arccosh
    float sind  = fmaxf(s2, 1e-12f);
    float coef  = dist / sind;
    float4 o;
    o.x = coef * (v.x + ((lane == 0) ? x0 : 0.0f));  // dim0: x0 + (inner/ref_sq)*1
    o.y = coef * v.y;
    o.z = coef * v.z;
    o.w = coef * v.w;
    *(float4*)(tangent + (size_t)wid * IN_DIM + d0) = o;
}

// ---------------------------------------------------------------------------
// 2) Per-node perspectives GEMM (fp32 WMMA 16x16x4) + gate softmax + combine.
//    Block = 256 threads = 8 waves; 16 nodes per block.
//    Wave w owns cols [w*64, w*64+64) of the 512-wide U = tan @ W_per.
// ---------------------------------------------------------------------------
__global__ __launch_bounds__(256) void gemm_gate_kernel(
    const float* __restrict__ tangent, const float* __restrict__ W_per,
    const float* __restrict__ b_per, const float* __restrict__ W_gate,
    const float* __restrict__ b_gate, float* __restrict__ u_hat, int n_nodes) {
    __shared__ float tanL[16][IN_DIM];
    __shared__ float uhatL[16][OUT_DIM];
    __shared__ float gateL[16][NUM_PERSP];

    int tid   = threadIdx.x;
    int lane  = tid & 31;
    int wave  = tid >> 5;
    int node0 = blockIdx.x * 16;

    for (int i = 0; i < 8; ++i) {               // 2048 floats, 8 per thread
        int idx = tid + i * 256;
        int r = idx >> 7, d = idx & 127;
        int n = node0 + r;
        tanL[r][d]  = (n < n_nodes) ? tangent[(size_t)n * IN_DIM + d] : 0.0f;
        uhatL[r][d] = 0.0f;
    }
    __syncthreads();

    v8f acc[4];
    for (int t = 0; t < 4; ++t)
        for (int j = 0; j < 8; ++j) acc[t][j] = 0.0f;

    const int arow = lane & 15;
    const int koff = (lane >> 4) << 1;          // 0 for lanes 0-15, 2 for 16-31
    const int colb = wave * 64;
    for (int kk = 0; kk < IN_DIM; kk += 4) {
        v2f a;
        a.x = tanL[arow][kk + koff];
        a.y = tanL[arow][kk + koff + 1];
        #pragma unroll
        for (int t = 0; t < 4; ++t) {
            int col = colb + t * 16 + (lane & 15);
            v2f b;
            b.x = W_per[(size_t)(kk + koff) * WCOLS + col];
            b.y = W_per[(size_t)(kk + koff + 1) * WCOLS + col];
            acc[t] = __builtin_amdgcn_wmma_f32_16x16x4_f32(
                false, a, false, b, (short)0, acc[t], false, false);
        }
    }

    // Gate logits: 16 nodes x 4 persp = 64 dot products of length 128
    if (tid < 64) {
        int r = tid >> 2, p = tid & 3;
        float s = b_gate[p];
        for (int k = 0; k < IN_DIM; ++k) s += tanL[r][k] * W_gate[k * NUM_PERSP + p];
        gateL[r][p] = s;
    }
    __syncthreads();
    if (tid < 16) {
        float g0 = gateL[tid][0], g1 = gateL[tid][1];
        float g2 = gateL[tid][2], g3 = gateL[tid][3];
        float mx = fmaxf(fmaxf(g0, g1), fmaxf(g2, g3));
        float e0 = __expf(g0 - mx), e1 = __expf(g1 - mx);
        float e2 = __expf(g2 - mx), e3 = __expf(g3 - mx);
        float zi = 1.0f / (e0 + e1 + e2 + e3);
        gateL[tid][0] = e0 * zi; gateL[tid][1] = e1 * zi;
        gateL[tid][2] = e2 * zi; gateL[tid][3] = e3 * zi;
    }
    __syncthreads();

    // Gated combine: u_hat[n,d] = sum_p gate[n,p] * (U[n, p*128+d] + b_per[p*128+d])
    #pragma unroll
    for (int t = 0; t < 4; ++t) {
        int cg = colb + t * 16 + (lane & 15);
        int p  = cg >> 7;
        int d  = cg & 127;
        float bp = b_per[cg];
        #pragma unroll
        for (int j = 0; j < 8; ++j) {
            int r = (lane < 16) ? j : (j + 8);
            atomicAdd(&uhatL[r][d], (acc[t][j] + bp) * gateL[r][p]);
        }
    }
    __syncthreads();
    for (int i = 0; i < 8; ++i) {
        int idx = tid + i * 256;
        int r = idx >> 7, d = idx & 127;
        int n = node0 + r;
        if (n < n_nodes) u_hat[(size_t)n * OUT_DIM + d] = uhatL[r][d];
    }
}

// ---------------------------------------------------------------------------
// Routing helpers
// ---------------------------------------------------------------------------
__global__ __launch_bounds__(256) void fill_b_kernel(float* __restrict__ b, int n) {
    int t = blockIdx.x * blockDim.x + threadIdx.x;
    if (t < n) b[t] = 0.0f;
}

__global__ __launch_bounds__(256) void fill_iter_kernel(
    float* __restrict__ m, float* __restrict__ z, float* __restrict__ s_raw,
    int n_nodes) {
    int t = blockIdx.x * blockDim.x + threadIdx.x;
    if (t < n_nodes * OUT_DIM) s_raw[t] = 0.0f;
    if (t < n_nodes) { m[t] = -INFINITY; z[t] = 0.0f; }
}

__global__ __launch_bounds__(256) void edge_max_kernel(
    const float* __restrict__ b_ij, const int* __restrict__ col,
    float* __restrict__ m, int n_edges) {
    int e = blockIdx.x * blockDim.x + threadIdx.x;
    if (e < n_edges) atomicMaxF(&m[col[e]], b_ij[e]);
}

// Fused: w = exp(b - m[col]); z[col] += w; s_raw[col,:] += w * (u_hat[row,:] + bias)
__global__ __launch_bounds__(256) void edge_scatter_kernel(
    const float* __restrict__ b_ij, const int* __restrict__ row,
    const int* __restrict__ col, const float* __restrict__ m,
    float* __restrict__ z, const float* __restrict__ u_hat,
    const float* __restrict__ bias, float* __restrict__ s_raw, int n_edges) {
    int wid  = (blockIdx.x * blockDim.x + threadIdx.x) >> 5;
    int lane = threadIdx.x & 31;
    if (wid >= n_edges) return;
    int c = col[wid], r = row[wid];
    float w = __expf(b_ij[wid] - m[c]);
    if (lane == 0) atomicAdd(&z[c], w);
    int d0 = lane * 4;
    float4 u  = *(const float4*)(u_hat + (size_t)r * OUT_DIM + d0);
    float4 bi = *(const float4*)(bias + d0);
    float* sp = s_raw + (size_t)c * OUT_DIM + d0;
    atomicAdd(sp + 0, w * (u.x + bi.x));
    atomicAdd(sp + 1, w * (u.y + bi.y));
    atomicAdd(sp + 2, w * (u.z + bi.z));
    atomicAdd(sp + 3, w * (u.w + bi.w));
}

// s = s_raw / z ; s_j = (|s|^2/(1+|s|^2)) * s / sqrt(|s|^2 + 1e-9)
__global__ __launch_bounds__(256) void squash_kernel(
    const float* __restrict__ s_raw, const float* __restrict__ z,
    float* __restrict__ s_j, int n_nodes) {
    int wid  = (blockIdx.x * blockDim.x + threadIdx.x) >> 5;
    int lane = threadIdx.x & 31;
    if (wid >= n_nodes) return;
    float zz = z[wid];
    int d0 = lane * 4;
    float4 s = make_float4(0.0f, 0.0f, 0.0f, 0.0f);
    if (zz > 0.0f) {
        s = *(const float4*)(s_raw + (size_t)wid * OUT_DIM + d0);
        float inv = 1.0f / zz;
        s.x *= inv; s.y *= inv; s.z *= inv; s.w *= inv;
    }
    float nsq = s.x * s.x + s.y * s.y + s.z * s.z + s.w * s.w;
    nsq = waveReduceSum(nsq);
    float f = (nsq / (1.0f + nsq)) / sqrtf(nsq + 1e-9f);
    float4 o = make_float4(f * s.x, f * s.y, f * s.z, f * s.w);
    *(float4*)(s_j + (size_t)wid * OUT_DIM + d0) = o;
}

// b_ij[e] += dot(s_j[col[e]], u_hat[row[e]])
__global__ __launch_bounds__(256) void bupdate_kernel(
    float* __restrict__ b_ij, const int* __restrict__ row,
    const int* __restrict__ col, const float* __restrict__ s_j,
    const float* __restrict__ u_hat, int n_edges) {
    int wid  = (blockIdx.x * blockDim.x + threadIdx.x) >> 5;
    int lane = threadIdx.x & 31;
    if (wid >= n_edges) return;
    int c = col[wid], r = row[wid];
    int d0 = lane * 4;
    float4 s = *(const float4*)(s_j + (size_t)c * OUT_DIM + d0);
    float4 u = *(const float4*)(u_hat + (size_t)r * OUT_DIM + d0);
    float acc = s.x * u.x + s.y * u.y + s.z * u.z + s.w * u.w;
    acc = waveReduceSum(acc);
    if (lane == 0) b_ij[wid] += acc;
}

// out = cosh(vn)*e0 + (sinh(vn)/vn)*v ; vn = min(sqrt(|inner(v,v)|+eps), 10)
__global__ __launch_bounds__(256) void expmap_kernel(
    const float* __restrict__ s_j, float* __restrict__ out, int n_nodes) {
    int wid  = (blockIdx.x * blockDim.x + threadIdx.x) >> 5;
    int lane = threadIdx.x & 31;
    if (wid >= n_nodes) return;
    int d0 = lane * 4;
    float4 v = *(const float4*)(s_j + (size_t)wid * OUT_DIM + d0);
    float ss = v.x * v.x + v.y * v.y + v.z * v.z + v.w * v.w;
    ss = waveReduceSum(ss);
    float v0 = __shfl(v.x, 0, 32);
    float inner = ss - 2.0f * v0 * v0;
    float vn = fminf(sqrtf(fabsf(inner) + 1e-12f), 10.0f);
    float sh = sinhf(vn) / vn;
    float ch = coshf(vn);
    float4 o = make_float4(sh * v.x, sh * v.y, sh * v.z, sh * v.w);
    if (lane == 0) o.x += ch;
    *(float4*)(out + (size_t)wid * OUT_DIM + d0) = o;
}

// ---------------------------------------------------------------------------
extern "C" void kernel_launch(void* const* d_in, const int* in_sizes, int n_in,
                              void* d_out, int out_size, void* d_ws, size_t ws_size,
                              hipStream_t stream) {
    const float* x      = (const float*)d_in[0];
    const int*   eidx   = (const int*)d_in[1];
    const float* W_per  = (const float*)d_in[2];
    const float* b_per  = (const float*)d_in[3];
    const float* W_gate = (const float*)d_in[4];
    const float* b_gate = (const float*)d_in[5];
    const float* bias   = (const float*)d_in[6];
    float* out = (float*)d_out;

    const int n_nodes = in_sizes[0] / IN_DIM;
    const int n_edges = in_sizes[1] / 2;
    const int* row = eidx;
    const int* col = eidx + n_edges;

    // Workspace layout (floats): u_hat | tangent(->s_raw) | s_j | m | z | b_ij
    float* ws      = (float*)d_ws;
    float* u_hat   = ws;
    float* tangent = ws + (size_t)n_nodes * OUT_DIM;        // reused as s_raw
    float* s_raw   = tangent;
    float* s_j     = ws + 2 * (size_t)n_nodes * OUT_DIM;
    float* m       = ws + 3 * (size_t)n_nodes * OUT_DIM;
    float* z       = m + n_nodes;
    float* b_ij    = z + n_nodes;
    // total: 3*N*128 + 2*N + E floats (~79 MB)

    const int nodeWaveBlocks = (n_nodes + 7) / 8;   // 8 waves (nodes) / 256-thr block
    const int edgeWaveBlocks = (n_edges + 7) / 8;

    fill_b_kernel<<<(n_edges + 255) / 256, 256, 0, stream>>>(b_ij, n_edges);
    tangent_kernel<<<nodeWaveBlocks, 256, 0, stream>>>(x, tangent, n_nodes);
    gemm_gate_kernel<<<(n_nodes + 15) / 16, 256, 0, stream>>>(
        tangent, W_per, b_per, W_gate, b_gate, u_hat, n_nodes);
    // NOTE: tangent buffer is dead from here on; it becomes s_raw.

    for (int r = 0; r < 3; ++r) {
        fill_iter_kernel<<<(n_nodes * OUT_DIM + 255) / 256, 256, 0, stream>>>(
            m, z, s_raw, n_nodes);
        edge_max_kernel<<<(n_edges + 255) / 256, 256, 0, stream>>>(
            b_ij, col, m, n_edges);
        edge_scatter_kernel<<<edgeWaveBlocks, 256, 0, stream>>>(
            b_ij, row, col, m, z, u_hat, bias, s_raw, n_edges);
        squash_kernel<<<nodeWaveBlocks, 256, 0, stream>>>(s_raw, z, s_j, n_nodes);
        if (r < 2)
            bupdate_kernel<<<edgeWaveBlocks, 256, 0, stream>>>(
                b_ij, row, col, s_j, u_hat, n_edges);
    }
    expmap_kernel<<<nodeWaveBlocks, 256, 0, stream>>>(s_j, out, n_nodes);
}